// FastIMDCT4_43258910605328
// MI455X (gfx1250) — compile-verified
//
#include <hip/hip_runtime.h>
#include <cstdint>

// ---------------------------------------------------------------------------
// FastIMDCT for MI455X (gfx1250, wave32, WMMA).
//
//   K1: per-frame 512-pt complex FFT via 16x32 four-step; DFT-16 / DFT-32
//       stages run on V_WMMA_F32_16X16X4_F32 chains (96 WMMA per frame).
//       Twiddles come from LDS tables built once per workgroup; the input
//       matrix M is stored transposed so every WMMA B-fragment is one
//       aligned ds_load_b64. Signal reads are non-temporal (read-once) so
//       the 192 MB L2 stays owned by the 128 MB inter buffer.
//   K2: overlap-add gather (each trimmed output = exactly 2 contributions);
//       inter reads hit L2, output stores are non-temporal (write-once).
//
// d_ws requirement: 32*1024*1024*4 = 134,217,728 bytes.
// ---------------------------------------------------------------------------

typedef float v2f __attribute__((ext_vector_type(2)));
typedef float v4f __attribute__((ext_vector_type(4)));
typedef float v8f __attribute__((ext_vector_type(8)));

static constexpr float TWO_PI = 6.28318530717958647692f;

static __device__ __forceinline__ v8f wmma4(v2f a, v2f b, v8f c) {
  // D = A(16x4,f32) * B(4x16,f32) + C. NEG on A/B must be 0 for f32
  // (ISA 7.12 NEG table), so negations are baked into operands.
  return __builtin_amdgcn_wmma_f32_16x16x4_f32(
      false, a, false, b, (short)0, c, false, false);
}

// ------------------------------- Kernel 1 ----------------------------------
// One wave per frame, 4 waves (128 threads) per block, 8192 blocks.
__global__ __launch_bounds__(128) void imdct_fft_kernel(
    const float* __restrict__ sig, float* __restrict__ inter) {
  // Twiddle tables (block-shared, frame-independent): {cos, sin} pairs.
  __shared__ __align__(16) float2 tb_tw[512];    // exp(-2pi*i*(k+0.125)/2048)
  __shared__ __align__(16) float2 tb_w512[512];  // [n2*16+k1] exp(-2pi*i*k1*n2/512)
  __shared__ __align__(16) float2 tb_f16[256];   // [m*16+j]   exp(-2pi*i*m*j/16)
  __shared__ __align__(16) float2 tb_g32[1024];  // [k2*32+n2] exp(-2pi*i*n2*k2/32)
  // Per-wave staging.
  __shared__ __align__(16) float lds_raw[4][1024];  // raw frame / inter stage
  __shared__ __align__(16) float lds_mr[4][512];    // Mt_re (32x16) / T_re (16x32)
  __shared__ __align__(16) float lds_mi[4][512];    // Mt_im (32x16) / T_im (16x32)

  const int tid  = threadIdx.x;
  const int wave = tid >> 5;
  const int lane = tid & 31;
  const int h    = lane >> 4;  // lane half (ISA fragment layouts)
  const int lm   = lane & 15;  // lane mod 16
  const int frame = blockIdx.x * 4 + wave;  // frame = b*1024 + t

  // ---- build twiddle tables once per block (~18 sincos per thread) -------
  for (int i = tid; i < 512; i += 128) {  // tw
    float sn, cs;
    __sincosf(-TWO_PI * ((float)i + 0.125f) * (1.0f / 2048.0f), &sn, &cs);
    tb_tw[i] = make_float2(cs, sn);
  }
  for (int i = tid; i < 512; i += 128) {  // W512: n2 = i>>4, k1 = i&15
    float sn, cs;
    __sincosf(-TWO_PI * (float)((i >> 4) * (i & 15)) * (1.0f / 512.0f), &sn, &cs);
    tb_w512[i] = make_float2(cs, sn);
  }
  for (int i = tid; i < 256; i += 128) {  // F16: m = i>>4, j = i&15
    float sn, cs;
    __sincosf(-TWO_PI * (float)((i >> 4) * (i & 15)) * (1.0f / 16.0f), &sn, &cs);
    tb_f16[i] = make_float2(cs, sn);
  }
  for (int i = tid; i < 1024; i += 128) {  // G32: k2 = i>>5, n2 = i&31
    float sn, cs;
    __sincosf(-TWO_PI * (float)((i >> 5) * (i & 31)) * (1.0f / 32.0f), &sn, &cs);
    tb_g32[i] = make_float2(cs, sn);
  }

  float* raw = lds_raw[wave];
  float* Mr  = lds_mr[wave];
  float* Mi  = lds_mi[wave];
  const float* src = sig + (size_t)frame * 1024;

  // ---- 1) coalesced non-temporal load of the frame into LDS --------------
#pragma unroll
  for (int m = 0; m < 8; ++m) {
    const int idx = lane * 4 + m * 128;
    const v4f d =
        __builtin_nontemporal_load(reinterpret_cast<const v4f*>(src + idx));
    *reinterpret_cast<v4f*>(raw + idx) = d;
  }
  __syncthreads();  // tables + raw ready

  // ---- 2) pre-permute + pre-twiddle: Mt[n2][n1] = tw[k]*c[k], k=32*n1+n2 -
  // Transposed store (n2 = lane, n1 = m): per-lane contiguous -> b128 stores,
  // and stage-A B-fragments become single aligned ds_load_b64.
#pragma unroll
  for (int mm = 0; mm < 4; ++mm) {
    v4f vr, vi;
#pragma unroll
    for (int j = 0; j < 4; ++j) {
      const int m = mm * 4 + j;
      const int k = lane + 32 * m;
      const float s0 = raw[2 * k];
      const float s1 = raw[1023 - 2 * k];
      const float2 w = tb_tw[k];
      vr[j] = w.x * s0 - w.y * s1;
      vi[j] = w.x * s1 + w.y * s0;
    }
    *reinterpret_cast<v4f*>(&Mr[lane * 16 + mm * 4]) = vr;
    *reinterpret_cast<v4f*>(&Mi[lane * 16 + mm * 4]) = vi;
  }
  __syncthreads();

  // ---- DFT-16 A-fragments from table: cols (4kc+2h, 4kc+2h+1) ------------
  v2f AFr[4], AFi[4], AFn[4];
#pragma unroll
  for (int kc = 0; kc < 4; ++kc) {
    const float4 f =
        *reinterpret_cast<const float4*>(&tb_f16[lm * 16 + 4 * kc + 2 * h]);
    AFr[kc] = (v2f){f.x, f.z};
    AFi[kc] = (v2f){f.y, f.w};
    AFn[kc] = (v2f){-f.y, -f.w};
  }

  // ---- 3) stage A: Y(16x32) = F16 * M (WMMA), then k1*n2 twiddle ---------
  // B-frag element (v,lane) = M[K=4kc+v+2h][N=col] = Mt[col*16 + 4kc+v+2h].
  v8f Yr[2], Yi[2];
#pragma unroll
  for (int t = 0; t < 2; ++t) {
    v8f yr = {};
    v8f yi = {};
    const int col = 16 * t + lm;  // n2
#pragma unroll
    for (int kc = 0; kc < 4; ++kc) {
      const int c0 = col * 16 + 4 * kc + 2 * h;  // even -> 8B aligned
      const v2f br = *reinterpret_cast<const v2f*>(&Mr[c0]);
      const v2f bi = *reinterpret_cast<const v2f*>(&Mi[c0]);
      yr = wmma4(AFr[kc], br, yr);  // +Fr*Mr
      yr = wmma4(AFn[kc], bi, yr);  // -Fi*Mi
      yi = wmma4(AFr[kc], bi, yi);  // +Fr*Mi
      yi = wmma4(AFi[kc], br, yi);  // +Fi*Mr
    }
    // twiddle exp(-2pi*i*k1*n2/512); C/D layout: k1 = v + 8h.
#pragma unroll
    for (int v = 0; v < 8; ++v) {
      const float2 w = tb_w512[col * 16 + v + 8 * h];
      Yr[t][v] = w.x * yr[v] - w.y * yi[v];
      Yi[t][v] = w.x * yi[v] + w.y * yr[v];
    }
  }
  __syncthreads();

  // ---- 4) scatter twiddled T back to LDS in (k1,n2) row-major ------------
#pragma unroll
  for (int t = 0; t < 2; ++t) {
#pragma unroll
    for (int v = 0; v < 8; ++v) {
      const int k1 = v + 8 * h;
      const int n2 = 16 * t + lm;
      Mr[k1 * 32 + n2] = Yr[t][v];
      Mi[k1 * 32 + n2] = Yi[t][v];
    }
  }
  __syncthreads();

  // ---- 5) stage C: Z(16x32) = T * G (DFT-32 over n2) ---------------------
  // A-frag element (v,lane) = T[k1=lm][n2=4kc+v+2h]: contiguous v2f.
#pragma unroll
  for (int tp = 0; tp < 2; ++tp) {
    v8f zr = {};
    v8f zi = {};
    const int k2 = lm + 16 * tp;
#pragma unroll
    for (int kc = 0; kc < 8; ++kc) {
      const int c0 = 4 * kc + 2 * h;  // even -> aligned v2f / float4
      const v2f ar = *reinterpret_cast<const v2f*>(&Mr[lm * 32 + c0]);
      const v2f ai = *reinterpret_cast<const v2f*>(&Mi[lm * 32 + c0]);
      const float4 g =
          *reinterpret_cast<const float4*>(&tb_g32[k2 * 32 + c0]);
      const v2f gr = (v2f){g.x, g.z};
      const v2f gi = (v2f){g.y, g.w};
      const v2f gn = (v2f){-g.y, -g.w};
      zr = wmma4(ar, gr, zr);  // +Tr*Gr
      zr = wmma4(ai, gn, zr);  // -Ti*Gi
      zi = wmma4(ar, gi, zi);  // +Tr*Gi
      zi = wmma4(ai, gr, zi);  // +Ti*Gr
    }
    // ---- 6) post-twiddle tw[k], interleave re/im into LDS "inter" --------
#pragma unroll
    for (int v = 0; v < 8; ++v) {
      const int kk = (v + 8 * h) + 16 * k2;  // FFT bin index
      const float2 w = tb_tw[kk];
      raw[2 * kk]     = w.x * zr[v] - w.y * zi[v];
      raw[2 * kk + 1] = w.x * zi[v] + w.y * zr[v];
    }
  }
  __syncthreads();

  // ---- 7) coalesced store of inter frame (default RT: keep in L2) --------
  float* dst = inter + (size_t)frame * 1024;
#pragma unroll
  for (int m = 0; m < 8; ++m) {
    const int idx = lane * 4 + m * 128;
    *reinterpret_cast<v4f*>(dst + idx) =
        *reinterpret_cast<const v4f*>(raw + idx);
  }
}

// ------------------------------- Kernel 2 ----------------------------------
// idx[i] (i<1024): even->i, odd->1024-i ; idx[i>=1024] = idx[2047-i].
// post[j] = idx[(j+512)&2047] ; sign_rolled[j] from pre-roll parity/quadrant.
static __device__ __forceinline__ int post_idx(int j) {
  const int i = (j + 512) & 2047;
  const int ii = (i < 1024) ? i : (2047 - i);
  return (ii & 1) ? (1024 - ii) : ii;
}
static __device__ __forceinline__ float win_val(int j,
                                                const float* __restrict__ w) {
  const int i = (j + 512) & 2047;
  const float s = ((i & 1) ? -1.0f : 1.0f) * ((i < 512) ? -1.0f : 1.0f);
  return s * w[j] * (4.0f / 2048.0f);
}

__global__ __launch_bounds__(256) void ola_kernel(
    const float* __restrict__ inter, const float* __restrict__ window,
    float* __restrict__ out) {
  const int b = blockIdx.y;
  const int i4 = blockIdx.x * 256 + threadIdx.x;
  const int p = i4 * 4;    // trimmed output index (float4 aligned)
  const int q = p + 1024;  // untrimmed index
  const int t2 = q >> 10;  // 1..1023 -> t2-1 in 0..1022: both frames valid
  const int j2 = q & 1023; // 0..1020; +3 stays in-frame (1024 % 4 == 0)
  const float* f2 = inter + ((size_t)b * 1024 + t2) * 1024;
  const float* f1 = f2 - 1024;

  v4f r;
#pragma unroll
  for (int e = 0; e < 4; ++e) {
    const int jb = j2 + e;
    const int ja = jb + 1024;
    r[e] = win_val(ja, window) * f1[post_idx(ja)] +
           win_val(jb, window) * f2[post_idx(jb)];
  }
  // Write-once output: non-temporal store keeps L2 owned by inter.
  __builtin_nontemporal_store(
      r, reinterpret_cast<v4f*>(out + (size_t)b * 1047552 + p));
}

// ------------------------------- Launch ------------------------------------
extern "C" void kernel_launch(void* const* d_in, const int* in_sizes, int n_in,
                              void* d_out, int out_size, void* d_ws,
                              size_t ws_size, hipStream_t stream) {
  const float* sig    = (const float*)d_in[0];  // (32,1024,1024) f32
  const float* window = (const float*)d_in[1];  // (2048,) f32
  float* inter = (float*)d_ws;                  // 128 MB scratch
  float* out   = (float*)d_out;                 // (32,1,1,1047552) f32

  // K1: 32768 frames, one wave each, 4 waves/block (50 KB LDS/block).
  imdct_fft_kernel<<<dim3(8192), dim3(128), 0, stream>>>(sig, inter);
  // K2: 1047552/4 float4 per batch = 1023 blocks of 256; 32 batches.
  ola_kernel<<<dim3(1023, 32), dim3(256), 0, stream>>>(inter, window, out);
}